// SoftRankMDS_82609400971282
// MI455X (gfx1250) — compile-verified
//
#include <hip/hip_runtime.h>
#include <hip/hip_bf16.h>

typedef float v2f __attribute__((ext_vector_type(2)));
typedef float v8f __attribute__((ext_vector_type(8)));

#define NPTS 512

// ---------------------------------------------------------------------------
// Stage 1: pairwise distance matrix via V_WMMA_F32_16X16X4_F32.
// One wave (32 lanes) computes one 16x16 tile of D.
//   G = Z * Z^T  (M=N=512, K=2 padded to 4)
//   D[i,j] = d2 > 0 ? sqrt(d2) : 0,   d2 = |zi|^2 + |zj|^2 - 2*G[i,j]
//
// A-matrix 16x4 f32 layout (ISA 7.12.2): lanes 0-15 hold M=lane,
//   VGPR0 = K0 (x), VGPR1 = K1 (y); lanes 16-31 hold K2/K3 = zero pad.
// B-matrix 4x16 f32: row-K striped across lanes within a VGPR (symmetric
//   to A): lanes 0-15 VGPR0 = row K0 (col x), VGPR1 = row K1 (col y);
//   lanes 16-31 = rows K2/K3 = zero pad.
// C/D 16x16 f32: VGPR v -> M = v (lanes 0-15) / v+8 (lanes 16-31), N = lane%16.
// ---------------------------------------------------------------------------
__global__ __launch_bounds__(128) void SoftRankMDS_dist_wmma(
    const float* __restrict__ Z, float* __restrict__ D) {
  const int lane = threadIdx.x & 31;
  const int wave = blockIdx.x * (blockDim.x >> 5) + (threadIdx.x >> 5);
  const int ti = (wave >> 5) << 4;  // 32 tiles per dimension
  const int tj = (wave & 31) << 4;
  const int l16 = lane & 15;
  const bool hi = lane >= 16;

  const float ax = Z[2 * (ti + l16) + 0];
  const float ay = Z[2 * (ti + l16) + 1];
  v2f a;
  a.x = hi ? 0.0f : ax;  // K0 row (x) / K2 zero-pad
  a.y = hi ? 0.0f : ay;  // K1 row (y) / K3 zero-pad

  const float bx = Z[2 * (tj + l16) + 0];
  const float by = Z[2 * (tj + l16) + 1];
  v2f b;
  b.x = hi ? 0.0f : bx;
  b.y = hi ? 0.0f : by;

  v8f c = {};
  // (neg_a, A, neg_b, B, c_mod, C, reuse_a, reuse_b)
  c = __builtin_amdgcn_wmma_f32_16x16x4_f32(false, a, false, b, (short)0, c,
                                            false, false);

  const float sqj = bx * bx + by * by;  // |z_col|^2 for this lane's N=l16

#pragma unroll
  for (int v = 0; v < 8; ++v) {
    const int m = v + (hi ? 8 : 0);
    const float zx = Z[2 * (ti + m) + 0];
    const float zy = Z[2 * (ti + m) + 1];
    const float d2 = zx * zx + zy * zy + sqj - 2.0f * c[v];
    // Raw v_sqrt_f32 (no IEEE denormal pre-scale / Newton fixup chain):
    // inputs are O(1e-8..1e-2), far from denormal, and the consumer is an
    // approximate sigmoid built on v_rcp_f32 anyway.
    D[(ti + m) * NPTS + tj + l16] =
        d2 > 0.0f ? __builtin_amdgcn_sqrtf(d2) : 0.0f;
  }
}

// ---------------------------------------------------------------------------
// Stage 2: soft ranks. One block per row i. Transcendental-throughput bound:
//   out[i,j] = sum_k sigmoid(alpha*(d[i,j]-d[i,k]))
//            = sum_k 1 / (1 + exp2(t_k - t_j)),  t = alpha*log2(e)*d[i,:]
// Row is staged (pre-scaled) in LDS; t[k] reads are wave-uniform broadcasts.
// Uses raw v_exp_f32 / v_rcp_f32. Reads+rewrites its own row of `out`
// in-place (strictly read-before-write within the block, rows disjoint
// across blocks), so no workspace is needed.
// ---------------------------------------------------------------------------
__global__ __launch_bounds__(256) void SoftRankMDS_softrank(
    float* out, const float* __restrict__ alpha) {
  __shared__ float t[NPTS];
  const int i = blockIdx.x;
  const float aL2E = alpha[0] * 1.4426950408889634f;  // alpha * log2(e)

  for (int k = threadIdx.x; k < NPTS; k += 256)
    t[k] = out[i * NPTS + k] * aL2E;
  __syncthreads();

#pragma unroll
  for (int r = 0; r < 2; ++r) {
    const int j = threadIdx.x + r * 256;
    const float tj = t[j];
    float s = 0.0f;
#pragma unroll 8
    for (int k = 0; k < NPTS; ++k) {
      const float e = __builtin_amdgcn_exp2f(t[k] - tj);  // v_exp_f32
      s += __builtin_amdgcn_rcpf(1.0f + e);               // v_rcp_f32
    }
    out[i * NPTS + j] = s;  // safe: all reads of this row went through LDS
  }
}

extern "C" void kernel_launch(void* const* d_in, const int* in_sizes, int n_in,
                              void* d_out, int out_size, void* d_ws,
                              size_t ws_size, hipStream_t stream) {
  (void)in_sizes; (void)n_in; (void)d_ws; (void)ws_size; (void)out_size;
  const float* Z = (const float*)d_in[0];
  const float* alpha = (const float*)d_in[1];
  float* out = (float*)d_out;

  // 1024 tiles, one wave each: 256 blocks x 4 waves.
  SoftRankMDS_dist_wmma<<<256, 128, 0, stream>>>(Z, out);
  // One block per row.
  SoftRankMDS_softrank<<<NPTS, 256, 0, stream>>>(out, alpha);
}